// OHEMBCEWithLogitsLoss_68513318306136
// MI455X (gfx1250) — compile-verified
//
#include <hip/hip_runtime.h>
#include <stdint.h>

// OHEM BCE-with-logits: histogram-select top-k threshold on loss bit patterns,
// then masked sum. 4 streaming passes; WMMA f32 16x16x4 used as the wave-level
// sum-reduction engine in the final pass.

typedef float v2f __attribute__((ext_vector_type(2)));
typedef float v8f __attribute__((ext_vector_type(8)));

#define OHEM_BINS 4096

__device__ __forceinline__ float ohem_bce_loss(float x, float t) {
    // max(x,0) - x*t + log1p(exp(-|x|)); always >= 0
    return fmaxf(x, 0.0f) - x * t + log1pf(expf(-fabsf(x)));
}

__device__ __forceinline__ void ohem_hist_accum(uint32_t* sh, uint32_t uu, int level, uint32_t prefix) {
    if (level == 0) {
        atomicAdd(&sh[uu >> 20], 1u);
    } else if (level == 1) {
        if ((uu >> 20) == prefix) atomicAdd(&sh[(uu >> 8) & 0xFFFu], 1u);
    } else {
        if ((uu >> 8) == prefix) atomicAdd(&sh[uu & 0xFFu], 1u);
    }
}

__global__ void ohem_init_kernel(uint32_t* hist, uint32_t* stats, double* acc, uint32_t kkept) {
    int i = blockIdx.x * blockDim.x + threadIdx.x;
    if (i < OHEM_BINS) hist[i] = 0u;
    if (i == 0) {
        stats[0] = 0u;       // prefix (grows to full 32-bit threshold pattern)
        stats[1] = kkept;    // remaining needed within candidate set
        *acc = 0.0;
    }
}

__global__ void ohem_hist_kernel(const float* __restrict__ lg, const float* __restrict__ tg,
                                 uint32_t* __restrict__ lossu_out, const uint32_t* __restrict__ lossu_in,
                                 uint32_t* __restrict__ hist, const uint32_t* __restrict__ stats,
                                 int level, int n) {
    __shared__ uint32_t sh[OHEM_BINS];
    for (int i = threadIdx.x; i < OHEM_BINS; i += blockDim.x) sh[i] = 0u;
    __syncthreads();
    const uint32_t prefix = (level > 0) ? stats[0] : 0u;

    const int n4 = n >> 2;
    const int stride = (int)(gridDim.x * blockDim.x);
    for (int i = (int)(blockIdx.x * blockDim.x + threadIdx.x); i < n4; i += stride) {
        uint32_t u0, u1, u2, u3;
        if (lossu_in) {
            uint4 lu = ((const uint4*)lossu_in)[i];
            u0 = lu.x; u1 = lu.y; u2 = lu.z; u3 = lu.w;
        } else {
            float4 xv = ((const float4*)lg)[i];
            float4 tv = ((const float4*)tg)[i];
            u0 = __float_as_uint(ohem_bce_loss(xv.x, tv.x));
            u1 = __float_as_uint(ohem_bce_loss(xv.y, tv.y));
            u2 = __float_as_uint(ohem_bce_loss(xv.z, tv.z));
            u3 = __float_as_uint(ohem_bce_loss(xv.w, tv.w));
        }
        if (lossu_out) {
            uint4 lu; lu.x = u0; lu.y = u1; lu.z = u2; lu.w = u3;
            ((uint4*)lossu_out)[i] = lu;
        }
        ohem_hist_accum(sh, u0, level, prefix);
        ohem_hist_accum(sh, u1, level, prefix);
        ohem_hist_accum(sh, u2, level, prefix);
        ohem_hist_accum(sh, u3, level, prefix);
    }
    // tail (n not multiple of 4)
    const int base = n4 << 2;
    if (blockIdx.x == 0) {
        for (int i = base + (int)threadIdx.x; i < n; i += (int)blockDim.x) {
            uint32_t uu = lossu_in ? lossu_in[i]
                                   : __float_as_uint(ohem_bce_loss(lg[i], tg[i]));
            if (lossu_out) lossu_out[i] = uu;
            ohem_hist_accum(sh, uu, level, prefix);
        }
    }
    __syncthreads();
    for (int i = threadIdx.x; i < OHEM_BINS; i += blockDim.x) {
        uint32_t c = sh[i];
        if (c) atomicAdd(&hist[i], c);
    }
}

// Single block: descend bins from top, find crossing, update prefix/remaining, zero hist.
__global__ void ohem_scan_kernel(uint32_t* hist, uint32_t* stats, int bits, int bins) {
    if (threadIdx.x == 0) {
        uint64_t rem = stats[1];
        uint64_t accc = 0;
        for (int b = bins - 1; b >= 0; --b) {
            uint64_t c = hist[b];
            if (accc + c >= rem) {
                stats[0] = (stats[0] << bits) | (uint32_t)b;
                stats[1] = (uint32_t)(rem - accc);   // remaining inside selected bin
                break;
            }
            accc += c;
        }
    }
    __syncthreads();
    for (int i = threadIdx.x; i < OHEM_BINS; i += blockDim.x) hist[i] = 0u;
}

// Sum all loss values strictly above the 32-bit threshold pattern T.
// Wave reduction done with V_WMMA_F32_16X16X4_F32 (B = ones => D row m is row-sum of A).
__global__ void ohem_sum_kernel(const float* __restrict__ lg, const float* __restrict__ tg,
                                const uint32_t* __restrict__ lossu_in,
                                const uint32_t* __restrict__ stats, double* acc, int n) {
    const uint32_t T = stats[0];
    float p0 = 0.0f, p1 = 0.0f;

    const int n4 = n >> 2;
    const int stride = (int)(gridDim.x * blockDim.x);
    for (int i = (int)(blockIdx.x * blockDim.x + threadIdx.x); i < n4; i += stride) {
        uint32_t u0, u1, u2, u3;
        if (lossu_in) {
            uint4 lu = ((const uint4*)lossu_in)[i];
            u0 = lu.x; u1 = lu.y; u2 = lu.z; u3 = lu.w;
        } else {
            float4 xv = ((const float4*)lg)[i];
            float4 tv = ((const float4*)tg)[i];
            u0 = __float_as_uint(ohem_bce_loss(xv.x, tv.x));
            u1 = __float_as_uint(ohem_bce_loss(xv.y, tv.y));
            u2 = __float_as_uint(ohem_bce_loss(xv.z, tv.z));
            u3 = __float_as_uint(ohem_bce_loss(xv.w, tv.w));
        }
        p0 += (u0 > T) ? __uint_as_float(u0) : 0.0f;
        p1 += (u1 > T) ? __uint_as_float(u1) : 0.0f;
        p0 += (u2 > T) ? __uint_as_float(u2) : 0.0f;
        p1 += (u3 > T) ? __uint_as_float(u3) : 0.0f;
    }
    const int base = n4 << 2;
    if (blockIdx.x == 0) {
        for (int i = base + (int)threadIdx.x; i < n; i += (int)blockDim.x) {
            uint32_t uu = lossu_in ? lossu_in[i]
                                   : __float_as_uint(ohem_bce_loss(lg[i], tg[i]));
            p0 += (uu > T) ? __uint_as_float(uu) : 0.0f;
        }
    }

    // --- wave-level sum via WMMA f32 16x16x4: A[m][k] = 32 lanes' {p0,p1}, B = ones ---
    v2f a; a.x = p0; a.y = p1;
    v2f b; b.x = 1.0f; b.y = 1.0f;
    v8f c = {};
    v8f d = __builtin_amdgcn_wmma_f32_16x16x4_f32(false, a, false, b, (short)0, c, false, false);
    // Lane L<16 holds v_0..v_7 in d[0..7]; lane L>=16 holds v_8..v_15 (v_m = rowsum m).
    float s = d[0] + d[1] + d[2] + d[3] + d[4] + d[5] + d[6] + d[7];
    float tot = __uint_as_float((uint32_t)__builtin_amdgcn_readlane((int)__float_as_uint(s), 0)) +
                __uint_as_float((uint32_t)__builtin_amdgcn_readlane((int)__float_as_uint(s), 16));

    __shared__ float wsum[32];
    const int lane = (int)(threadIdx.x & 31u);
    const int wave = (int)(threadIdx.x >> 5u);
    if (lane == 0) wsum[wave] = tot;
    __syncthreads();
    if (threadIdx.x == 0) {
        double bs = 0.0;
        const int nw = (int)((blockDim.x + 31u) >> 5u);
        for (int w = 0; w < nw; ++w) bs += (double)wsum[w];
        atomicAdd(acc, bs);
    }
}

__global__ void ohem_finalize_kernel(const uint32_t* stats, const double* acc, float* out, uint32_t kkept) {
    if (blockIdx.x == 0 && threadIdx.x == 0) {
        const double s = *acc;
        const float tval = __uint_as_float(stats[0]);
        const double rem = (double)stats[1];
        out[0] = (float)((s + rem * (double)tval) / (double)kkept);
    }
}

extern "C" void kernel_launch(void* const* d_in, const int* in_sizes, int n_in,
                              void* d_out, int out_size, void* d_ws, size_t ws_size,
                              hipStream_t stream) {
    const float* lg = (const float*)d_in[0];
    const float* tg = (const float*)d_in[1];
    float* out = (float*)d_out;
    const int n = in_sizes[0];

    long long kept = (long long)((double)n * 0.25);
    if (kept < 10000) kept = 10000;
    if (kept > n) kept = n;
    const uint32_t kkept = (uint32_t)kept;

    uint8_t* ws = (uint8_t*)d_ws;
    uint32_t* hist = (uint32_t*)ws;                              // 16 KB
    uint32_t* stats = (uint32_t*)(ws + OHEM_BINS * 4);           // 16 B
    double* acc = (double*)(ws + OHEM_BINS * 4 + 16);            // 8 B
    const size_t loss_off = 32768;
    uint32_t* lossu = nullptr;                                   // stored loss bit patterns (optional)
    if (ws_size >= loss_off + (size_t)n * sizeof(uint32_t))
        lossu = (uint32_t*)(ws + loss_off);

    const int threads = 256;
    int n4 = n >> 2; if (n4 < 1) n4 = 1;
    int blocks = (n4 + threads - 1) / threads;
    if (blocks > 4096) blocks = 4096;
    if (blocks < 1) blocks = 1;

    ohem_init_kernel<<<dim3((OHEM_BINS + 255) / 256), dim3(256), 0, stream>>>(hist, stats, acc, kkept);

    // Level 0: bits [31:20], also persist loss patterns if workspace allows.
    ohem_hist_kernel<<<dim3(blocks), dim3(threads), 0, stream>>>(lg, tg, lossu, nullptr, hist, stats, 0, n);
    ohem_scan_kernel<<<dim3(1), dim3(256), 0, stream>>>(hist, stats, 12, 4096);

    // Level 1: bits [19:8] within prefix.
    ohem_hist_kernel<<<dim3(blocks), dim3(threads), 0, stream>>>(lg, tg, nullptr, lossu, hist, stats, 1, n);
    ohem_scan_kernel<<<dim3(1), dim3(256), 0, stream>>>(hist, stats, 12, 4096);

    // Level 2: bits [7:0] within prefix -> exact 32-bit threshold.
    ohem_hist_kernel<<<dim3(blocks), dim3(threads), 0, stream>>>(lg, tg, nullptr, lossu, hist, stats, 2, n);
    ohem_scan_kernel<<<dim3(1), dim3(256), 0, stream>>>(hist, stats, 8, 256);

    // Masked sum above threshold (WMMA wave reduction), then finalize.
    ohem_sum_kernel<<<dim3(blocks), dim3(threads), 0, stream>>>(lg, tg, lossu, stats, acc, n);
    ohem_finalize_kernel<<<dim3(1), dim3(1), 0, stream>>>(stats, acc, out, kkept);
}